// HierarchicalLoss_66425964199957
// MI455X (gfx1250) — compile-verified
//
#include <hip/hip_runtime.h>

// HierarchicalLoss on gfx1250 (MI455X):
//   loss = mean_p[ 3*logZ - log S8(t) - log S4(t) - (l_t - m) ] + ln8 + ln4
// Pure streaming reduction: 151 MB traffic -> ~6.5us at 23.3 TB/s.
// b128 loads (4 pixels/thread), wave32 sum via V_WMMA_F32_16X16X4_F32,
// deterministic two-pass block-sum reduction (no atomics).

typedef __attribute__((ext_vector_type(2))) float v2f;
typedef __attribute__((ext_vector_type(8))) float v8f;

#define NUM_C 16
#define HW    262144      // 512*512
#define NPIX  2097152     // 8*512*512
#define NQUAD 524288      // NPIX/4
#define NBLK  512
#define NTHR  256

__device__ __forceinline__ float comp4(float4 v, int j) {
  return j == 0 ? v.x : j == 1 ? v.y : j == 2 ? v.z : v.w;  // j is constant after unroll
}

// Wave32 sum-reduction via WMMA: A(16x4) = per-lane value in col {0,2}, zeros
// in col {1,3}; B(4x16) = ones. D[m,n] = v_m + v_(m+16) for all n.
// Lane L<16 holds D[0..7,L], lane L>=16 holds D[8..15,L-16]; summing each
// lane's 8 D regs and xor-16 combining gives the full 32-lane sum.
__device__ __forceinline__ float wave_sum_wmma(float v) {
  v2f a; a[0] = v;    a[1] = 0.0f;
  v2f b; b[0] = 1.0f; b[1] = 1.0f;
  v8f c = {};
  c = __builtin_amdgcn_wmma_f32_16x16x4_f32(false, a, false, b, (short)0, c,
                                            false, false);
  float s = c[0] + c[1] + c[2] + c[3] + c[4] + c[5] + c[6] + c[7];
  s += __shfl_xor(s, 16, 32);
  return s;
}

__global__ __launch_bounds__(NTHR) void hloss_main(
    const float* __restrict__ logits,
    const long long* __restrict__ target,
    float* __restrict__ blocksums) {
  const int tid = blockIdx.x * NTHR + threadIdx.x;
  const int nthreads = NBLK * NTHR;
  float vacc = 0.0f;

  for (int q = tid; q < NQUAD; q += nthreads) {
    const int p4 = q << 2;                 // first pixel of this quad
    const int b  = p4 >> 18;               // p4 / (512*512)
    const int hw = p4 & (HW - 1);
    const float* base = logits + (size_t)b * (NUM_C * HW) + hw;

    // Prefetch next grid-stride tile (speculative; OOB is silently dropped).
    const int qn = q + nthreads;
    if (qn < NQUAD) {
      const int p4n = qn << 2;
      const float* nb =
          logits + (size_t)(p4n >> 18) * (NUM_C * HW) + (p4n & (HW - 1));
      __builtin_prefetch(nb, 0, 3);
      __builtin_prefetch(nb + 8 * HW, 0, 3);
    }

    // 16 x global_load_b128: 4 consecutive pixels, all 16 channels.
    float4 l4[NUM_C];
#pragma unroll
    for (int c = 0; c < NUM_C; ++c)
      l4[c] = *(const float4*)(base + c * HW);

    int t[4];
#pragma unroll
    for (int j = 0; j < 4; ++j) t[j] = (int)target[p4 + j];

#pragma unroll
    for (int j = 0; j < 4; ++j) {
      float m = comp4(l4[0], j);
#pragma unroll
      for (int c = 1; c < NUM_C; ++c) m = fmaxf(m, comp4(l4[c], j));

      float s4g[4] = {0.f, 0.f, 0.f, 0.f};
      float lt = 0.0f;
#pragma unroll
      for (int c = 0; c < NUM_C; ++c) {
        const float lc = comp4(l4[c], j);
        const float ec = __expf(lc - m);
        s4g[c >> 2] += ec;                 // constant index after unroll
        lt = (t[j] == c) ? lc : lt;        // v_cndmask select of target logit
      }
      const float s8lo = s4g[0] + s4g[1];
      const float s8hi = s4g[2] + s4g[3];
      const float Z    = s8lo + s8hi;

      const int g4 = t[j] >> 2;
      float s4t = s4g[0];
      s4t = (g4 == 1) ? s4g[1] : s4t;
      s4t = (g4 == 2) ? s4g[2] : s4t;
      s4t = (g4 == 3) ? s4g[3] : s4t;
      const float s8t = (t[j] >> 3) ? s8hi : s8lo;

      vacc += 3.0f * __logf(Z) - __logf(s8t) - __logf(s4t) - (lt - m);
    }
  }

  // Wave reduction (EXEC is all-ones here: uniform control flow, 8 full waves).
  const float wsum = wave_sum_wmma(vacc);

  __shared__ float wsums[NTHR / 32];
  const int lane = threadIdx.x & 31;
  const int wid  = threadIdx.x >> 5;
  if (lane == 0) wsums[wid] = wsum;
  __syncthreads();
  if (threadIdx.x == 0) {
    float bs = 0.0f;
#pragma unroll
    for (int i = 0; i < NTHR / 32; ++i) bs += wsums[i];
    blocksums[blockIdx.x] = bs;            // deterministic: one writer per slot
  }
}

__global__ __launch_bounds__(256) void hloss_final(
    const float* __restrict__ blocksums, float* __restrict__ out) {
  __shared__ double sh[256];
  double s = 0.0;
  for (int i = threadIdx.x; i < NBLK; i += 256) s += (double)blocksums[i];
  sh[threadIdx.x] = s;
  __syncthreads();
  for (int off = 128; off > 0; off >>= 1) {
    if (threadIdx.x < off) sh[threadIdx.x] += sh[threadIdx.x + off];
    __syncthreads();
  }
  if (threadIdx.x == 0) {
    // + ln(8) + ln(4) = ln(32)
    out[0] = (float)(sh[0] * (1.0 / (double)NPIX) + 3.4657359027997265);
  }
}

extern "C" void kernel_launch(void* const* d_in, const int* in_sizes, int n_in,
                              void* d_out, int out_size, void* d_ws,
                              size_t ws_size, hipStream_t stream) {
  const float*     logits = (const float*)d_in[0];      // [8,16,512,512] f32
  const long long* target = (const long long*)d_in[1];  // [8,512,512] i64
  float* blocksums = (float*)d_ws;                      // NBLK floats (2 KB)
  float* out       = (float*)d_out;                     // scalar f32

  hloss_main<<<NBLK, NTHR, 0, stream>>>(logits, target, blocksums);
  hloss_final<<<1, 256, 0, stream>>>(blocksums, out);
}